// PsActivation_31774168055919
// MI455X (gfx1250) — compile-verified
//
#include <hip/hip_runtime.h>
#include <stdint.h>

#define NUMH 4096
#define KCOLS 9
#define NB    8192            // quantizer buckets
#define PSZ   8448            // padded P entries (>= NB+1, multiple of 4 for b128 fill)

typedef float f4 __attribute__((ext_vector_type(4)));

// Identical quantizer used by table-build and apply kernels (monotone in v).
__device__ __forceinline__ int ps_bucket(float v, float e0, float invw) {
  return (int)floorf((v - e0) * invw);
}

// ---------------------------------------------------------------------------
// Kernel 1: collapse the 8-step threshold loop into per-row value F[i], and
// emit contiguous copy of sorted column h[:,0]. Branch-free.
// ---------------------------------------------------------------------------
__global__ __launch_bounds__(256) void ps_precompute(
    const float* __restrict__ h, const float* __restrict__ d,
    const float* __restrict__ T, const float* __restrict__ b,
    float* __restrict__ Hcol, float* __restrict__ F) {
  int i = blockIdx.x * blockDim.x + threadIdx.x;
  if (i >= NUMH) return;
  float dv[9], Tv[9];
#pragma unroll
  for (int t = 1; t <= 8; ++t) { dv[t] = d[t]; Tv[t] = T[t]; }
  const float* row = h + i * KCOLS;
  float h0  = row[0];
  float acc = -b[0];
  acc += ((h0 - Tv[1]) >= 0.0f) ? dv[1] : 0.0f;     // t=1 uses column 0
#pragma unroll
  for (int t = 2; t <= 8; ++t)                      // t=2..8 use column t
    acc += ((row[t] - Tv[t]) >= 0.0f) ? dv[t] : 0.0f;
  Hcol[i] = h0;
  F[i]    = acc;
}

// first index idx with ps_bucket(Hcol[idx]) >= b  (== count of f < b)
__device__ __forceinline__ int ps_lbound_bucket(const float* __restrict__ Hcol,
                                                float e0, float invw, int b) {
  int lo = 0, hi = NUMH;
#pragma unroll
  for (int k = 0; k < 12; ++k) {
    int mid = (lo + hi) >> 1;
    if (ps_bucket(Hcol[mid], e0, invw) < b) lo = mid + 1; else hi = mid;
  }
  return lo;
}

// ---------------------------------------------------------------------------
// Kernel 1b: build fused neighbor records S[i] = {h[i-1], h[i], F[i-1], F[i]}
// and bucket range pairs P[b] = (count(f<b)) | (count(f<=b) << 16).
// ---------------------------------------------------------------------------
__global__ __launch_bounds__(256) void ps_tables(
    const float* __restrict__ Hcol, const float* __restrict__ F,
    f4* __restrict__ S, uint32_t* __restrict__ P) {
  int i = blockIdx.x * blockDim.x + threadIdx.x;
  if (i < NUMH) {
    int im1 = (i > 0) ? i - 1 : 0;
    f4 s;
    s.x = Hcol[im1]; s.y = Hcol[i];
    s.z = F[im1];    s.w = F[i];
    S[i] = s;
  }
  if (i <= NB) {
    float e0   = Hcol[0];
    float em   = Hcol[NUMH - 1];
    float invw = (float)NB / (em - e0);
    int lo1 = ps_lbound_bucket(Hcol, e0, invw, i);
    int lo2 = ps_lbound_bucket(Hcol, e0, invw, i + 1);
    P[i] = (uint32_t)lo1 | ((uint32_t)lo2 << 16);
  }
}

// ---------------------------------------------------------------------------
// Kernel 2: streaming apply. Tables staged into LDS with CDNA5 async LDS
// loads (ASYNCcnt); per element: bucket -> tiny lower_bound -> one b128.
// ---------------------------------------------------------------------------
__global__ __launch_bounds__(256) void ps_apply(
    const f4* __restrict__ x, const float* __restrict__ Hcol,
    const f4* __restrict__ S, const uint32_t* __restrict__ P,
    f4* __restrict__ out, int n4) {
  __shared__ f4       sS[NUMH];    // 64 KB
  __shared__ uint32_t sP[PSZ];     // 33 KB
  const int tid = threadIdx.x;

  // ---- async LDS fill (b128 per lane, ASYNCcnt-tracked) ----
  {
    for (int e = tid; e < NUMH; e += 256) {
      unsigned           l = (unsigned)(size_t)&sS[e];
      unsigned long long g = (unsigned long long)(size_t)&S[e];
      asm volatile("global_load_async_to_lds_b128 %0, %1, off"
                   :: "v"(l), "v"(g) : "memory");
    }
    const f4* P4 = reinterpret_cast<const f4*>(P);
    for (int e = tid; e < PSZ / 4; e += 256) {
      unsigned           l = (unsigned)(size_t)&sP[e * 4];
      unsigned long long g = (unsigned long long)(size_t)&P4[e];
      asm volatile("global_load_async_to_lds_b128 %0, %1, off"
                   :: "v"(l), "v"(g) : "memory");
    }
    asm volatile("s_wait_asynccnt 0" ::: "memory");
  }
  __syncthreads();

  const float e0   = Hcol[0];
  const float em   = Hcol[NUMH - 1];
  const float invw = (float)NB / (em - e0);
  const float* sSf = (const float*)sS;

  const int stride = gridDim.x * blockDim.x;
  for (int i = blockIdx.x * blockDim.x + tid; i < n4; i += stride) {
    __builtin_prefetch(&x[i + 2 * stride], 0, 0);     // global_prefetch_b8
    f4 v = __builtin_nontemporal_load(&x[i]);
    f4 r;
#pragma unroll
    for (int c = 0; c < 4; ++c) {
      float xv = v[c];
      int b = ps_bucket(xv, e0, invw);
      b = b < 0 ? 0 : (b > NB ? NB : b);
      uint32_t pr = sP[b];
      int lo = (int)(pr & 0xFFFFu);
      int hi = (int)(pr >> 16);
      while (lo < hi) {                               // usually 0-2 iterations
        int mid = (lo + hi) >> 1;
        if (sSf[mid * 4 + 1] < xv) lo = mid + 1; else hi = mid;
      }
      int idx = lo < 1 ? 1 : (lo > NUMH - 1 ? NUMH - 1 : lo);
      f4 s = sS[idx];                                 // {left,right,F_l,F_r}
      r[c] = (fabsf(xv - s.x) < fabsf(xv - s.y)) ? s.z : s.w;
    }
    __builtin_nontemporal_store(r, &out[i]);
  }
}

// ---------------------------------------------------------------------------
extern "C" void kernel_launch(void* const* d_in, const int* in_sizes, int n_in,
                              void* d_out, int out_size, void* d_ws, size_t ws_size,
                              hipStream_t stream) {
  const float* x = (const float*)d_in[0];   // (8,4096,4096) f32
  const float* h = (const float*)d_in[1];   // (4096,9) f32
  const float* d = (const float*)d_in[2];   // (9,) f32
  const float* T = (const float*)d_in[3];   // (9,) f32
  const float* b = (const float*)d_in[4];   // scalar f32

  // workspace layout (16B aligned): F | Hcol | S | P
  float*    F    = (float*)d_ws;                         // 16 KB
  float*    Hcol = F + NUMH;                             // 16 KB
  f4*       S    = (f4*)(Hcol + NUMH);                   // 64 KB
  uint32_t* P    = (uint32_t*)(S + NUMH);                // 33 KB (PSZ entries)

  ps_precompute<<<(NUMH + 255) / 256, 256, 0, stream>>>(h, d, T, b, Hcol, F);
  ps_tables<<<(NB + 1 + 255) / 256, 256, 0, stream>>>(Hcol, F, S, P);

  int n4 = in_sizes[0] / 4;                 // 33,554,432 float4s (divides exactly)
  ps_apply<<<2048, 256, 0, stream>>>((const f4*)x, Hcol, S, P, (f4*)d_out, n4);
}